// GridGCNNearNeighbors_33698313404549
// MI455X (gfx1250) — compile-verified
//
#include <hip/hip_runtime.h>

// Ball query (PointNet++ query_ball_point): B=8, N=8192, S=2048, R=0.2, K=32.
//
// MI455X plan:
//  * Stage the batch's point slab (8192*3 f32 = 96 KB) into LDS once per block
//    via the Tensor Data Mover (tensor_load_to_lds + s_wait_tensorcnt).
//  * Distance matrix via V_WMMA_F32_16X16X4_F32:
//      A row m = [cx, cy, cz, |c|^2]   (16 centroids x K4)
//      B col n = [-2px, -2py, -2pz, 1] (K4 x 16 points)
//      C[m][n] = |c|^2 - 2 c.p  ->  hit iff C <= R^2 - |p|^2 (per-lane thr)
//  * Ordered top-32 selection from wave32 ballots, uniform early exit.

typedef float v2f  __attribute__((ext_vector_type(2)));
typedef float v8f  __attribute__((ext_vector_type(8)));
typedef unsigned int u32x4 __attribute__((ext_vector_type(4)));
typedef int   i32x4 __attribute__((ext_vector_type(4)));
typedef int   i32x8 __attribute__((ext_vector_type(8)));

#define NPTS    8192
#define SCENT   2048
#define NBATCH  8
#define NNB     32
#define RADIUS2 0.04f
#define TILE_S  16          // centroids per wave
#define WPB     8           // waves per block (256 threads)

__device__ __forceinline__ unsigned int ballot32(bool pred) {
  return __builtin_amdgcn_ballot_w32(pred);
}

// Issue a TDM DMA of `nElemsF32` floats from `gaddr` into LDS offset `ldsOff`.
// 1-D descriptor: data_size=4B, tensor_dim0=tile_dim0=nElems, groups 2/3 zero.
__device__ __forceinline__ void tdm_load_f32(unsigned ldsOff,
                                             unsigned long long gaddr,
                                             unsigned nElemsF32) {
  u32x4 g0;
  g0[0] = 1u;                               // count=1, is_restore=0, gather=0
  g0[1] = ldsOff;                           // lds_addr [63:32]
  g0[2] = (unsigned)(gaddr & 0xFFFFFFFFull);        // global_addr [95:64]
  g0[3] = (unsigned)((gaddr >> 32) & 0x01FFFFFFull) // global_addr [120:96]
          | 0x80000000u;                            // type=2 ("image") [127:126]

  i32x8 g1;
  g1[0] = (int)(2u << 16);                  // wg_mask=0, data_size=2 (4 bytes)
  g1[1] = (int)((nElemsF32 & 0xFFFFu) << 16);       // tensor_dim0[15:0] @ [63:48]
  g1[2] = (int)(((nElemsF32 >> 16) & 0xFFFFu)       // tensor_dim0[31:16] @ [79:64]
          | (1u << 16));                            // tensor_dim1=1 @ [95:80]
  g1[3] = (int)((nElemsF32 & 0xFFFFu) << 16);       // tile_dim0 @ [127:112]
  g1[4] = 0;                                // tile_dim1=0, tile_dim2=0
  g1[5] = (int)nElemsF32;                   // tensor_dim0_stride[31:0]
  g1[6] = 0;                                // stride[47:32], dim1_stride lo
  g1[7] = 0;

  i32x4 z = {0, 0, 0, 0};
#if defined(__clang_major__) && (__clang_major__ >= 23)
  i32x8 z8 = {0, 0, 0, 0, 0, 0, 0, 0};
  __builtin_amdgcn_tensor_load_to_lds(g0, g1, z, z, z8, 0);
#else
  __builtin_amdgcn_tensor_load_to_lds(g0, g1, z, z, 0);
#endif
}

__global__ __launch_bounds__(256) void GridGCNNearNeighbors_kernel(
    const float* __restrict__ pos,       // (B, N, 3) f32
    const long long* __restrict__ cent,  // (B, S) i64
    int* __restrict__ out)               // (B, S, 32) i32
{
  __shared__ float spos[NPTS * 3];       // 96 KB: whole batch's point slab

  const int lane = threadIdx.x & 31;
  const int wave = threadIdx.x >> 5;
  const int tile = blockIdx.x * WPB + wave;           // global centroid tile
  const int tilesPerBatch = SCENT / TILE_S;           // 128 (8 tiles/block -> same b)
  const int b  = tile / tilesPerBatch;
  const int s0 = (tile % tilesPerBatch) * TILE_S;

  const float* posB = pos + (size_t)b * NPTS * 3;

  // ---- TDM: DMA the 96 KB point slab into LDS (wave 0 issues, all sync) ----
  if (wave == 0) {
    tdm_load_f32((unsigned)(unsigned long long)(uintptr_t)&spos[0],
                 (unsigned long long)(uintptr_t)posB, NPTS * 3);
    __builtin_amdgcn_s_wait_tensorcnt(0);
  }
  __syncthreads();

  const int m = lane & 15;               // slot within tile (mirrored halves)
  const bool lo = (lane < 16);

  // ---- build A (centroids) from LDS: lane<16 -> {cx,cy}; lane>=16 -> {cz,|c|^2}
  const long long ci = cent[(size_t)b * SCENT + s0 + m];
  const float cx = spos[ci * 3 + 0];
  const float cy = spos[ci * 3 + 1];
  const float cz = spos[ci * 3 + 2];
  const float cn = cx * cx + cy * cy + cz * cz;

  v2f A;
  A.x = lo ? cx : cz;
  A.y = lo ? cy : cn;

  const int cidx = lane & 7;
  int cnt   = 0;
  int first = NPTS;

  for (int n0 = 0; n0 < NPTS; n0 += 16) {
    // ---- build B (points) from LDS: lane<16 -> {-2px,-2py}; lane>=16 -> {-2pz,1}
    const float px = spos[(n0 + m) * 3 + 0];
    const float py = spos[(n0 + m) * 3 + 1];
    const float pz = spos[(n0 + m) * 3 + 2];
    const float pn = px * px + py * py + pz * pz;

    v2f Bm;
    Bm.x = lo ? (-2.0f * px) : (-2.0f * pz);
    Bm.y = lo ? (-2.0f * py) : 1.0f;

    v8f C = {};
    C = __builtin_amdgcn_wmma_f32_16x16x4_f32(
        /*neg_a=*/false, A, /*neg_b=*/false, Bm,
        /*c_mod=*/(short)0, C, /*reuse_a=*/false, /*reuse_b=*/false);

    // hit iff C[v] <= R^2 - |p|^2 ; C VGPR v = centroids (v, v+8), point = lane%16
    const float thr = RADIUS2 - pn;
    const unsigned bl0 = ballot32(C[0] <= thr);
    const unsigned bl1 = ballot32(C[1] <= thr);
    const unsigned bl2 = ballot32(C[2] <= thr);
    const unsigned bl3 = ballot32(C[3] <= thr);
    const unsigned bl4 = ballot32(C[4] <= thr);
    const unsigned bl5 = ballot32(C[5] <= thr);
    const unsigned bl6 = ballot32(C[6] <= thr);
    const unsigned bl7 = ballot32(C[7] <= thr);

    // each lane selects its centroid's 16-bit ordered hit-mask (branch-free)
    unsigned msk = bl0;
    msk = (cidx == 1) ? bl1 : msk;
    msk = (cidx == 2) ? bl2 : msk;
    msk = (cidx == 3) ? bl3 : msk;
    msk = (cidx == 4) ? bl4 : msk;
    msk = (cidx == 5) ? bl5 : msk;
    msk = (cidx == 6) ? bl6 : msk;
    msk = (cidx == 7) ? bl7 : msk;
    msk = (lane & 8) ? (msk >> 16) : (msk & 0xFFFFu);

    // ordered append of hit indices (lanes 16..31 mirror, writes guarded)
    while (msk != 0u && cnt < NNB) {
      const int t    = __builtin_ctz(msk);
      const int pidx = n0 + t;
      if (cnt == 0) first = pidx;
      if (lo) out[((size_t)b * SCENT + s0 + m) * NNB + cnt] = pidx;
      ++cnt;
      msk &= (msk - 1u);
    }

    // wave-uniform early exit once every centroid in the tile has 32 hits
    if (ballot32(cnt < NNB) == 0u) break;
  }

  // pad short lists with the first hit (or N when no hit, matching reference)
  if (lo) {
    const int fill = first;   // == NPTS iff no neighbor found
    for (int k = cnt; k < NNB; ++k)
      out[((size_t)b * SCENT + s0 + m) * NNB + k] = fill;
  }
}

extern "C" void kernel_launch(void* const* d_in, const int* in_sizes, int n_in,
                              void* d_out, int out_size, void* d_ws, size_t ws_size,
                              hipStream_t stream) {
  (void)in_sizes; (void)n_in; (void)out_size; (void)d_ws; (void)ws_size;
  const float*     pos  = (const float*)d_in[0];      // (8, 8192, 3) f32
  const long long* cent = (const long long*)d_in[1];  // (8, 2048) i64
  int*             out  = (int*)d_out;                // (8, 2048, 32) i32

  const int blocks = NBATCH * SCENT / (TILE_S * WPB); // 128 blocks x 256 threads
  GridGCNNearNeighbors_kernel<<<dim3(blocks), dim3(256), 0, stream>>>(pos, cent, out);
}